// NTMReadHead_1503238553612
// MI455X (gfx1250) — compile-verified
//
#include <hip/hip_runtime.h>
#include <math.h>
#include <stdint.h>

typedef float v2f __attribute__((ext_vector_type(2)));
typedef float v8f __attribute__((ext_vector_type(8)));

// Match the builtin's parameter pointee type exactly (from hipcc diagnostic):
//   "__attribute__((__vector_size__(4 * sizeof(int)))) int __device__ *"
typedef int async_v4i __attribute__((vector_size(4 * sizeof(int))));
typedef __attribute__((address_space(1))) async_v4i* g_v4i_p;   // global (AS1)
typedef __attribute__((address_space(3))) async_v4i* l_v4i_p;   // LDS (AS3)

#define CTRL 512
#define NDIM 128
#define MDIM 64
#define ODIM 70
#define EPSF 1e-16f
#define TSTRIDE 68   // LDS tile row stride (floats): 272B, 16B-aligned rows

#if __has_builtin(__builtin_amdgcn_global_load_async_to_lds_b128) && \
    __has_builtin(__builtin_amdgcn_s_wait_asynccnt)
#define USE_ASYNC_LDS 1
#else
#define USE_ASYNC_LDS 0
#endif

// Issue one async 16-byte global->LDS copy (ASYNCcnt-tracked DMA; no VGPR
// round-trip). AS1 ptr == flat global address; AS3 ptr == low 32 bits of the
// flat LDS address, so integer casts are
// sufficient and avoid addrspacecast legality questions.
__device__ __forceinline__ void async_b128(const float* gsrc, float* ldst) {
#if USE_ASYNC_LDS
    __builtin_amdgcn_global_load_async_to_lds_b128(
        (g_v4i_p)(uintptr_t)gsrc,
        (l_v4i_p)(uint32_t)(uintptr_t)ldst,
        0, 0);
#else
    const float4 v = *(const float4*)gsrc;
    float* d = ldst;
    d[0] = v.x; d[1] = v.y; d[2] = v.z; d[3] = v.w;
#endif
}

__device__ __forceinline__ void async_wait_all() {
#if USE_ASYNC_LDS
    __builtin_amdgcn_s_wait_asynccnt(0);
#endif
}

// ---------------------------------------------------------------------------
// Kernel 1: o = emb @ W^T + b   (16384x512 @ 512x70) via V_WMMA_F32_16X16X4_F32
// grid = B/16 blocks, block = 160 threads (5 waves). Wave w owns output cols
// [16w, 16w+16); cols >= 70 computed on a clamped W row and masked at store.
// A tile (16 x 512 f32) staged in LDS via async B128 (stride 516: rows are
// 2064B = 129*16 -> every B128 write 16B-aligned; A-frag b64 reads conflict-
// free). B fragment read from the L2-resident 140KB W.
// ---------------------------------------------------------------------------
__global__ __launch_bounds__(160) void ctrl_gemm_wmma(
    const float* __restrict__ emb, const float* __restrict__ W,
    const float* __restrict__ bias, float* __restrict__ o)
{
    __shared__ float sA[16 * 516];
    const int tid     = threadIdx.x;
    const int lane    = tid & 31;
    const int wave    = tid >> 5;
    const int rowBase = blockIdx.x * 16;

    // stage A tile: 16 rows x 512 f32 (async global->LDS, 2048 B128 packets)
    for (int f = tid; f < 16 * (CTRL / 4); f += 160) {
        const int r  = f >> 7;            // f / 128
        const int c4 = (f & 127) << 2;    // (f % 128) * 4
        async_b128(emb + (size_t)(rowBase + r) * CTRL + c4, sA + r * 516 + c4);
    }
    async_wait_all();
    __syncthreads();

    // Fragment addressing (wave32):
    //  A 16x4 f32: lane l -> row M = l&15, K = k0 + 2*(l>>4) + {0,1}
    //  B 4x16 f32: lane l -> col N = l&15, K = k0 + 2*(l>>4) + {0,1}
    const int mrow  = lane & 15;
    const int khalf = (lane >> 4) << 1;          // 0 or 2
    const int col   = wave * 16 + mrow;
    const int colc  = (col < ODIM) ? col : (ODIM - 1);  // clamp: in-bounds loads
    const float* wrow = W + (size_t)colc * CTRL;
    const float* arow = sA + mrow * 516;

    v8f acc = {};
    #pragma unroll 4
    for (int k0 = 0; k0 < CTRL; k0 += 4) {
        const int c = k0 + khalf;
        v2f a, bf;
        a.x  = arow[c];
        a.y  = arow[c + 1];
        bf.x = wrow[c];
        bf.y = wrow[c + 1];
        // (neg_a, A, neg_b, B, c_mod, C, reuse_a, reuse_b)
        acc = __builtin_amdgcn_wmma_f32_16x16x4_f32(false, a, false, bf,
                                                    (short)0, acc, false, false);
    }

    // D layout: VGPR v -> M = v + 8*(lane>>4), N = lane&15
    if (col < ODIM) {
        const float bv = bias[col];
        const int mOff = (lane >> 4) * 8;
        #pragma unroll
        for (int v = 0; v < 8; ++v) {
            const int row = rowBase + mOff + v;
            o[(size_t)row * ODIM + col] = acc[v] + bv;
        }
    }
}

// ---------------------------------------------------------------------------
// Kernel 2: fused NTM addressing + read. One block (128 threads, 4 waves) per
// batch item. memory[b] (128x64 f32 = 32KB) staged ONCE into LDS via async
// B128 DMA (stride 68 -> 16B-aligned rows; dot pass has a benign 2-way bank
// conflict, read pass conflict-free). HBM sees memory exactly once
// (~512MB total -> ~23us at 23.3TB/s).
// ---------------------------------------------------------------------------
__global__ __launch_bounds__(128) void ntm_address_read(
    const float* __restrict__ o,       // (B, 70)
    const float* __restrict__ w_prev,  // (B, 128)
    const float* __restrict__ memory,  // (B, 128, 64)
    float* __restrict__ out_r,         // (B, 64)
    float* __restrict__ out_w)         // (B, 128)
{
    __shared__ float tile[NDIM * TSTRIDE];
    __shared__ float sk[MDIM];
    __shared__ float sv[NDIM];
    __shared__ float sw[NDIM];
    __shared__ float sscal[8];   // beta, g, s0, s1, s2, gamma, normK

    const int tid = threadIdx.x;
    const int b   = blockIdx.x;
    const float* ob   = o + (size_t)b * ODIM;
    const float* memb = memory + (size_t)b * NDIM * MDIM;

    // Kick off the tile DMA FIRST so it overlaps the scalar/activation work.
    for (int f = tid; f < NDIM * (MDIM / 4); f += 128) {   // 2048 B128 packets
        const int r = f >> 4;
        const int c = (f & 15) << 2;
        async_b128(memb + r * MDIM + c, tile + r * TSTRIDE + c);
    }

    if (tid < MDIM) sk[tid] = ob[tid] + EPSF;
    if (tid == 0) {
        const float x = ob[64];
        sscal[0] = fmaxf(x, 0.f) + log1pf(expf(-fabsf(x)));          // beta
        sscal[1] = 1.f / (1.f + expf(-ob[65]));                      // g
        const float a0 = ob[66], a1 = ob[67], a2 = ob[68];
        const float mx = fmaxf(a0, fmaxf(a1, a2));
        const float e0 = expf(a0 - mx), e1 = expf(a1 - mx), e2 = expf(a2 - mx);
        const float es = e0 + e1 + e2;
        sscal[2] = e0 / es; sscal[3] = e1 / es; sscal[4] = e2 / es;  // s
        const float y = ob[69];
        sscal[5] = 1.f + fmaxf(y, 0.f) + log1pf(expf(-fabsf(y)));    // gamma
    }
    __syncthreads();

    if (tid == 0) {
        float s = 0.f;
        for (int m = 0; m < MDIM; ++m) s += sk[m] * sk[m];
        sscal[6] = sqrtf(s);                                         // ||k+eps||
    }
    async_wait_all();          // this wave's DMA packets landed in LDS
    __syncthreads();           // ... and everyone else's too

    const float beta = sscal[0], g = sscal[1];
    const float s0 = sscal[2], s1 = sscal[3], s2 = sscal[4];
    const float gamma = sscal[5], nK = sscal[6];
    const float wp = w_prev[(size_t)b * NDIM + tid];

    // cosine similarity per row n = tid (mem_eps vs k_eps)
    {
        const float* row = tile + tid * TSTRIDE;
        float dot = 0.f, nr = 0.f;
        #pragma unroll 8
        for (int m = 0; m < MDIM; ++m) {
            const float v = row[m] + EPSF;
            dot += v * sk[m];
            nr  += v * v;
        }
        sv[tid] = beta * (dot / (sqrtf(nr) * nK));
    }
    __syncthreads();

    // softmax over N = 128 (LDS-broadcast loops)
    float amax = -INFINITY;
    for (int n = 0; n < NDIM; ++n) amax = fmaxf(amax, sv[n]);
    const float e = expf(sv[tid] - amax);
    __syncthreads();
    sv[tid] = e;
    __syncthreads();
    float esum = 0.f;
    for (int n = 0; n < NDIM; ++n) esum += sv[n];

    const float wc = e / esum;
    const float wg = g * wc + (1.f - g) * wp;
    sw[tid] = wg;
    __syncthreads();

    // circular shift:  w_s[n] = s0*w_g[n-1] + s1*w_g[n] + s2*w_g[n+1]
    const float ws = s0 * sw[(tid + NDIM - 1) & (NDIM - 1)]
                   + s1 * sw[tid]
                   + s2 * sw[(tid + 1) & (NDIM - 1)];
    const float wsh = powf(ws, gamma);
    __syncthreads();
    sv[tid] = wsh;
    __syncthreads();
    float shsum = 0.f;
    for (int n = 0; n < NDIM; ++n) shsum += sv[n];

    const float w = wsh / (shsum + EPSF);
    sw[tid] = w;
    out_w[(size_t)b * NDIM + tid] = w;
    __syncthreads();

    // read vector: r[m] = sum_n w[n] * memory[b][n][m]  (raw memory, no eps)
    if (tid < MDIM) {
        float r = 0.f;
        #pragma unroll 8
        for (int n = 0; n < NDIM; ++n)
            r += sw[n] * tile[n * TSTRIDE + tid];
        out_r[(size_t)b * MDIM + tid] = r;
    }
}

// ---------------------------------------------------------------------------
extern "C" void kernel_launch(void* const* d_in, const int* in_sizes, int n_in,
                              void* d_out, int out_size, void* d_ws, size_t ws_size,
                              hipStream_t stream) {
    const float* emb    = (const float*)d_in[0];   // (B, 512)
    const float* w_prev = (const float*)d_in[1];   // (B, 128)
    const float* memory = (const float*)d_in[2];   // (B, 128, 64)
    const float* W      = (const float*)d_in[3];   // (70, 512)
    const float* bias   = (const float*)d_in[4];   // (70,)
    float* out = (float*)d_out;                    // r (B*64) then w (B*128)
    float* o   = (float*)d_ws;                     // (B, 70) controller outputs

    const int B = in_sizes[0] / CTRL;              // 16384

    ctrl_gemm_wmma<<<B / 16, 160, 0, stream>>>(emb, W, bias, o);
    ntm_address_read<<<B, 128, 0, stream>>>(o, w_prev, memory,
                                            out, out + (size_t)B * MDIM);
}